// RoIAlign_13082470383790
// MI455X (gfx1250) — compile-verified
//
#include <hip/hip_runtime.h>

// RoIAlign for MI455X (gfx1250): double-buffered TDM tile staging + LDS bilinear pooling.
// feat: (B=4, C=256, H=100, W=152) f32; boxes: (N,4); box_indices: (N)
// out: (N, 256, 7, 7) f32.

#define RA_P     7
#define RA_SR    2
#define RA_S     14          // RA_P * RA_SR sample points per axis
#define RA_C     256
#define RA_H     100
#define RA_W     152
#define RA_SCALE 0.25f
#define TILE_MAX 42          // max ROI span: 160 img px * 0.25 -> 40 feat px (+2 guard)
#define CCHUNK   4           // channels per TDM tile (2 buffers -> 56.4 KB LDS)
#define CB       64          // channels per block
#define NCHUNK   (CB / CCHUNK)
#define TILE_N   (CCHUNK * TILE_MAX * TILE_MAX)

#if defined(__has_builtin)
#if __has_builtin(__builtin_amdgcn_tensor_load_to_lds) && \
    __has_builtin(__builtin_amdgcn_s_wait_tensorcnt)
#define RA_USE_TDM 1
#endif
#endif
#ifndef RA_USE_TDM
#define RA_USE_TDM 0
#endif

typedef __attribute__((ext_vector_type(4))) unsigned int ra_u32x4;
typedef __attribute__((ext_vector_type(8))) int          ra_i32x8;
typedef __attribute__((ext_vector_type(4))) int          ra_i32x4;

#if RA_USE_TDM
// Issue one TDM 3D-tile load: feat[b, c0:c0+CCHUNK, y0:y0+Ht, x0:x0+Wt] -> LDS buffer.
// Must be called by a fully-active wave (uses readfirstlane); caller guards to wave 0.
__device__ __forceinline__ void ra_issue_tdm(const float* feat, int b, int c0,
                                             int y0, int x0, int Wt, int Ht,
                                             const float* lds_buf)
{
    const unsigned long long gaddr = (unsigned long long)(const void*)
        (feat + (((size_t)b * RA_C + (size_t)c0) * RA_H + (size_t)y0) * RA_W + (size_t)x0);

    const int wS  = __builtin_amdgcn_readfirstlane(Wt);
    const int hS  = __builtin_amdgcn_readfirstlane(Ht);
    const int glo = __builtin_amdgcn_readfirstlane((int)(unsigned)gaddr);
    const int ghi = __builtin_amdgcn_readfirstlane((int)(unsigned)(gaddr >> 32));
    const int lds = __builtin_amdgcn_readfirstlane((int)(unsigned)(size_t)(const void*)lds_buf);

    // D# group 0: count=1 | lds_addr | global_addr[56:0] | type=2
    ra_u32x4 g0;
    g0[0] = 1u;                                            // count=1, user mode
    g0[1] = (unsigned)lds;                                 // lds_addr (bytes)
    g0[2] = (unsigned)glo;                                 // global_addr[31:0]
    g0[3] = ((unsigned)ghi & 0x01FFFFFFu) | 0x80000000u;   // addr[56:32] | type=2

    // D# group 1: data_size=4B; tensor dims = tile dims (tile is fully in-bounds);
    // dim0 stride = W, dim1 stride (channel step) = H*W.
    ra_i32x8 g1;
    g1[0] = 0x20000;                                       // data_size=2 (4B), wg_mask=0
    g1[1] = (wS & 0xFFFF) << 16;                           // tensor_dim0[15:0]
    g1[2] = (int)(((unsigned)wS >> 16) | (((unsigned)hS & 0xFFFFu) << 16));
    g1[3] = (int)(((unsigned)hS >> 16) | (((unsigned)wS & 0xFFFFu) << 16)); // tile_dim0
    g1[4] = (hS & 0xFFFF) | (CCHUNK << 16);                // tile_dim1 | tile_dim2
    g1[5] = RA_W;                                          // tensor_dim0_stride[31:0]
    g1[6] = (int)((((unsigned)(RA_H * RA_W)) & 0xFFFFu) << 16); // dim1_stride[15:0]
    g1[7] = (RA_H * RA_W) >> 16;                           // dim1_stride[47:16]

    ra_i32x4 g2;                                           // tensor_dim2 = CCHUNK
    g2[0] = CCHUNK; g2[1] = 0; g2[2] = 0; g2[3] = 0;
    ra_i32x4 g3;
    g3[0] = 0; g3[1] = 0; g3[2] = 0; g3[3] = 0;
    ra_i32x8 g4;                                           // clang-23 extra group (unused)
    g4[0] = 0; g4[1] = 0; g4[2] = 0; g4[3] = 0;
    g4[4] = 0; g4[5] = 0; g4[6] = 0; g4[7] = 0;

    __builtin_amdgcn_tensor_load_to_lds(g0, g1, g2, g3, g4, 0);
}
#endif

__global__ __launch_bounds__(256) void roialign_tdm_kernel(
    const float* __restrict__ feat,
    const float* __restrict__ boxes,
    const int*   __restrict__ bidx,
    float*       __restrict__ out)
{
    // Separable per-ROI sample metadata: axis 0 = y, axis 1 = x.
    __shared__ int   s_lo[2][RA_S];
    __shared__ int   s_hi[2][RA_S];
    __shared__ float s_wl[2][RA_S];       // (1-frac) * valid  (low corner weight)
    __shared__ float s_wh[2][RA_S];       // frac * valid      (high corner weight)
    __shared__ float s_tile[2][TILE_N];   // double buffer, 2 x 28.2 KB

    const int n     = blockIdx.x;
    const int cbase = blockIdx.y * CB;
    const int tid   = threadIdx.x;

    if (tid < 2 * RA_S) {
        const int   axis = tid / RA_S;              // 0=y, 1=x
        const int   i    = tid - axis * RA_S;
        const float lim  = axis ? (float)RA_W : (float)RA_H;
        // boxes layout: x1,y1,x2,y2
        const float c1 = boxes[n * 4 + (axis ? 0 : 1)] * RA_SCALE;
        const float c2 = boxes[n * 4 + (axis ? 2 : 3)] * RA_SCALE;
        const float sz  = fmaxf(c2 - c1, 1.0f);
        const float pos = c1 + (sz / (float)RA_P) * (((float)i + 0.5f) / (float)RA_SR);
        const float valid = (pos > -1.0f && pos < lim) ? 1.0f : 0.0f;
        const float pc = fminf(fmaxf(pos, 0.0f), lim - 1.0f);
        const int   lo = (int)floorf(pc);
        const float l  = pc - (float)lo;
        s_lo[axis][i] = lo;
        s_hi[axis][i] = min(lo + 1, (int)lim - 1);
        s_wl[axis][i] = (1.0f - l) * valid;
        s_wh[axis][i] = l * valid;
    }
    __syncthreads();

    // Sample coords are monotonic -> tile bounds from first/last entries.
    const int y0 = s_lo[0][0];
    const int x0 = s_lo[1][0];
    const int Ht = min(s_hi[0][RA_S - 1] - y0 + 1, TILE_MAX);
    const int Wt = min(s_hi[1][RA_S - 1] - x0 + 1, TILE_MAX);
    const int b  = bidx[n];

#if RA_USE_TDM
    if ((tid >> 5) == 0) {   // wave 0 only (uniform branch; TDM ignores EXEC)
        ra_issue_tdm(feat, b, cbase, y0, x0, Wt, Ht, &s_tile[0][0]);
    }
#endif

    for (int chunk = 0; chunk < NCHUNK; ++chunk) {
        const int c0  = cbase + chunk * CCHUNK;
        const int buf = chunk & 1;

#if RA_USE_TDM
        if ((tid >> 5) == 0) {
            if (chunk + 1 < NCHUNK) {
                // Prefetch next tile into the other buffer, then wait only for
                // the current one (TDM ops from one wave complete in order).
                ra_issue_tdm(feat, b, c0 + CCHUNK, y0, x0, Wt, Ht, &s_tile[buf ^ 1][0]);
                __builtin_amdgcn_s_wait_tensorcnt(1);
            } else {
                __builtin_amdgcn_s_wait_tensorcnt(0);
            }
        }
#else
        // Fallback: cooperative tile load (also used by host-side parse).
        {
            const int tileN = CCHUNK * Ht * Wt;
            const float* src = feat +
                (((size_t)b * RA_C + (size_t)c0) * RA_H + (size_t)y0) * RA_W + (size_t)x0;
            for (int i = tid; i < tileN; i += 256) {
                const int cc = i / (Ht * Wt);
                const int r  = i - cc * (Ht * Wt);
                const int yy = r / Wt;
                const int xx = r - yy * Wt;
                s_tile[buf][i] = src[((size_t)cc * RA_H + (size_t)yy) * RA_W + (size_t)xx];
            }
        }
#endif
        __syncthreads();   // tile `chunk` visible to all waves

        // CCHUNK channels x 49 bins; idx-major keeps output stores contiguous.
        for (int idx = tid; idx < CCHUNK * 49; idx += 256) {
            const int cc  = idx / 49;
            const int bin = idx - cc * 49;
            const int by  = bin / 7;
            const int bx  = bin - by * 7;
            const float* t = &s_tile[buf][cc * Ht * Wt];

            float acc = 0.0f;
#pragma unroll
            for (int j = 0; j < RA_SR; ++j) {
                const int   sy  = by * RA_SR + j;
                const int   yl  = s_lo[0][sy] - y0;
                const int   yh  = s_hi[0][sy] - y0;
                const float wyl = s_wl[0][sy];
                const float wyh = s_wh[0][sy];
#pragma unroll
                for (int k = 0; k < RA_SR; ++k) {
                    const int   sx  = bx * RA_SR + k;
                    const int   xl  = s_lo[1][sx] - x0;
                    const int   xh  = s_hi[1][sx] - x0;
                    const float wxl = s_wl[1][sx];
                    const float wxh = s_wh[1][sx];
                    acc += wyl * (wxl * t[yl * Wt + xl] + wxh * t[yl * Wt + xh])
                         + wyh * (wxl * t[yh * Wt + xl] + wxh * t[yh * Wt + xh]);
                }
            }
            // Streaming (non-temporal) store: keep the L2 for the feature map.
            __builtin_nontemporal_store(acc * 0.25f,
                &out[((size_t)n * RA_C + (size_t)(c0 + cc)) * 49 + (size_t)bin]);
        }
        __syncthreads();   // all reads of buf done before it is overwritten
    }
}

extern "C" void kernel_launch(void* const* d_in, const int* in_sizes, int n_in,
                              void* d_out, int out_size, void* d_ws, size_t ws_size,
                              hipStream_t stream) {
    (void)n_in; (void)out_size; (void)d_ws; (void)ws_size;
    const float* feat  = (const float*)d_in[0];
    const float* boxes = (const float*)d_in[1];
    const int*   bidx  = (const int*)d_in[2];
    float*       out   = (float*)d_out;

    const int N = in_sizes[2];          // number of ROIs
    if (N <= 0) return;

    dim3 grid((unsigned)N, RA_C / CB, 1);   // (ROI, channel-slice)
    roialign_tdm_kernel<<<grid, 256, 0, stream>>>(feat, boxes, bidx, out);
}